// GaugeFluidNet_31954556682586
// MI455X (gfx1250) — compile-verified
//
#include <hip/hip_runtime.h>
#include <math.h>
#include <stdint.h>

// ---------------- problem constants ----------------
#define NB      32768
#define HID     64
#define DIM     8
#define LAYERS  6
#define INDIM   117
#define KPAD    120          // lift K padded to multiple of 4
#define BT      64           // batch rows per workgroup
#define NTHREADS 512         // 16 waves (wave32)

// LDS layout: h[blade][chan][row], chan pitch 66, plane skew +8
#define CPITCH  66
#define PLANE   (HID*CPITCH + 8)     // 4232 floats per blade plane
#define IDXH(k,c,b) ((k)*PLANE + (c)*CPITCH + (b))

typedef __attribute__((ext_vector_type(2))) float v2f;
typedef __attribute__((ext_vector_type(8))) float v8f;

// ---------------- Cl(3,0) geometric product table ----------------
// basis order [1, e1, e2, e3, e12, e13, e23, e123] as bitmasks {0,1,2,4,3,5,6,7}
struct GPTab { int idx[8][8]; float sgn[8][8]; };
constexpr int GMASK[8] = {0,1,2,4,3,5,6,7};
constexpr int GINV[8]  = {0,1,2,4,3,5,6,7};   // mask -> index (involution)
constexpr int cpopc(int x){ int c=0; while(x){ c += (x&1); x >>= 1; } return c; }
constexpr GPTab makeTab(){
  GPTab t{};
  for (int i=0;i<8;i++) for (int j=0;j<8;j++){
    int ma = GMASK[i], mb = GMASK[j];
    int a = ma >> 1, s = 0;
    while (a){ s += cpopc(a & mb); a >>= 1; }
    t.idx[i][j] = GINV[ma ^ mb];
    t.sgn[i][j] = (s & 1) ? -1.0f : 1.0f;
  }
  return t;
}
constexpr GPTab GP = makeTab();

// out = rotor(left, sparse blades {0,4,5,6}) * h
__device__ __forceinline__ void gp_rotor_left(float r0, float r4, float r5, float r6,
                                              const float* __restrict__ h,
                                              float* __restrict__ o) {
  #pragma unroll
  for (int k=0;k<8;k++) o[k] = 0.0f;
  const float rv[4] = {r0, r4, r5, r6};
  const int   ri[4] = {0, 4, 5, 6};
  #pragma unroll
  for (int ii=0; ii<4; ++ii) {
    #pragma unroll
    for (int j=0; j<8; ++j) {
      o[GP.idx[ri[ii]][j]] += GP.sgn[ri[ii]][j] * rv[ii] * h[j];
    }
  }
}
// out = h * rotor(right, sparse blades {0,4,5,6})
__device__ __forceinline__ void gp_rotor_right(const float* __restrict__ h,
                                               float r0, float r4, float r5, float r6,
                                               float* __restrict__ o) {
  #pragma unroll
  for (int k=0;k<8;k++) o[k] = 0.0f;
  const float rv[4] = {r0, r4, r5, r6};
  const int   rj[4] = {0, 4, 5, 6};
  #pragma unroll
  for (int i=0; i<8; ++i) {
    #pragma unroll
    for (int jj=0; jj<4; ++jj) {
      o[GP.idx[i][rj[jj]]] += GP.sgn[i][rj[jj]] * h[i] * rv[jj];
    }
  }
}

// ---------------- fused network kernel ----------------
__global__ __launch_bounds__(NTHREADS)
void gaugefluid_fused(const float* __restrict__ coords,      // [NB,5]
                      const float* __restrict__ tfreq,       // [16]
                      const float* __restrict__ rfreq,       // [16]
                      const float* __restrict__ lift_W,      // [117,512]
                      const float* __restrict__ lift_b,      // [512]
                      const float* __restrict__ in_ln_g,     // [64]
                      const float* __restrict__ ln_g,        // [6,64]
                      const float* __restrict__ rotor_biv,   // [6,64,3]
                      const float* __restrict__ act_w,       // [6,64]
                      const float* __restrict__ act_b,       // [6,64]
                      const float* __restrict__ lin_W,       // [6,8,64,64]
                      const float* __restrict__ lin_b,       // [6,64,8]
                      const float* __restrict__ out_ln_g,    // [64]
                      const float* __restrict__ blade_sel,   // [64,8]
                      const float* __restrict__ proj_W,      // [8,1,64]
                      const float* __restrict__ proj_b,      // [1,8]
                      float* __restrict__ out_final,         // [NB,8]
                      float* __restrict__ out_inter)         // [6,NB,64,8]
{
  __shared__ float shCur[8*PLANE];       // 135,424 B : residual / h
  __shared__ float shTmp[8*PLANE];       // 135,424 B : GEMM A operand / scratch
  __shared__ float shFeat[KPAD*CPITCH];  //  31,680 B : lift features, [K][row]
  __shared__ float shCoord[BT*5];

  const int tid  = threadIdx.x;
  const int lane = tid & 31;
  const int wave = tid >> 5;
  const int ln15 = lane & 15;
  const int hi   = lane >> 4;            // 0 or 1 (half-wave)
  const int B0   = blockIdx.x * BT;

  // ---- stage coords via async global->LDS DMA (ASYNCcnt path) ----
  if (tid < BT*5) {
    const unsigned ldsa = (unsigned)(uintptr_t)&shCoord[tid];
    const unsigned voff = (unsigned)(tid * 4);
    asm volatile("global_load_async_to_lds_b32 %0, %1, %2"
                 :: "v"(ldsa), "v"(voff), "s"(coords + (size_t)B0*5) : "memory");
  }
  asm volatile("s_wait_asynccnt 0x0" ::: "memory");
  __syncthreads();

  // ---- positional features -> shFeat[K][row] (rows 117..119 zero) ----
  {
    const int b  = tid & 63;
    const int fg = tid >> 6;
    float cc[5];
    #pragma unroll
    for (int q=0;q<5;q++) cc[q] = shCoord[b*5+q];
    for (int f = fg; f < KPAD; f += 8) {
      float v;
      if (f < 5) v = cc[f];
      else if (f < 53) {
        int fi = f - 5; int ci = fi >> 4; int r = fi & 15;
        float p = cc[ci] * (float)((r & 7) + 1);
        v = (r < 8) ? sinf(p) : cosf(p);
      } else if (f < 85) {
        int ti = f - 53; float p = cc[3] * tfreq[ti & 15];
        v = (ti < 16) ? sinf(p) : cosf(p);
      } else if (f < 117) {
        int ri = f - 85; float p = cc[4] * rfreq[ri & 15];
        v = (ri < 16) ? sinf(p) : cosf(p);
      } else v = 0.0f;
      shFeat[f*CPITCH + b] = v;
    }
  }
  __syncthreads();

  // ---- lift GEMM: feats[64x120] @ lift_W[120x512] -> shCur (fp32 WMMA) ----
  {
    for (int ntq = 0; ntq < 2; ++ntq) {
      const int nt = wave*2 + ntq;          // 0..31
      const int j  = nt*16 + ln15;          // output column 0..511
      const float bias = lift_b[j];
      for (int mt = 0; mt < 4; ++mt) {
        v8f acc;
        #pragma unroll
        for (int r = 0; r < 8; ++r) acc[r] = bias;
        const int bA = mt*16 + ln15;
        // kk = 0..28: all rows < 117, affine addresses, no clamping
        for (int kk = 0; kk < 29; ++kk) {
          const int i0 = kk*4 + 2*hi;
          v2f a, bf;
          a.x = shFeat[i0*CPITCH + bA];
          a.y = shFeat[(i0+1)*CPITCH + bA];
          bf.x = lift_W[(size_t)i0*512 + j];
          bf.y = lift_W[(size_t)(i0+1)*512 + j];
          acc = __builtin_amdgcn_wmma_f32_16x16x4_f32(false, a, false, bf,
                                                      (short)0, acc, false, false);
        }
        { // kk = 29 tail: A rows 116..119 (117..119 are zero); B rows clamped in-bounds
          const int i0 = 116 + 2*hi;
          v2f a, bf;
          a.x = shFeat[i0*CPITCH + bA];
          a.y = shFeat[(i0+1)*CPITCH + bA];
          bf.x = lift_W[(size_t)116*512 + j];   // only pairs with real A row 116 (hi=0)
          bf.y = bf.x;                          // pairs with zero A rows -> value irrelevant
          acc = __builtin_amdgcn_wmma_f32_16x16x4_f32(false, a, false, bf,
                                                      (short)0, acc, false, false);
        }
        #pragma unroll
        for (int r = 0; r < 8; ++r) {
          const int b = mt*16 + r + 8*hi;
          shCur[IDXH(j & 7, j >> 3, b)] = acc[r];
        }
      }
    }
  }
  __syncthreads();

  // ---- input clifford-LN (in place) ----
  {
    const int b  = tid & 63;
    const int cg = tid >> 6;
    for (int ci = 0; ci < 8; ++ci) {
      const int c = cg*8 + ci;
      float v[8], ss = 0.0f;
      #pragma unroll
      for (int k = 0; k < 8; ++k) { v[k] = shCur[IDXH(k,c,b)]; ss += v[k]*v[k]; }
      const float inv = in_ln_g[c] * rsqrtf(ss*0.125f + 1e-6f);
      #pragma unroll
      for (int k = 0; k < 8; ++k) shCur[IDXH(k,c,b)] = v[k]*inv;
    }
  }
  __syncthreads();

  // ================= layer loop =================
  for (int l = 0; l < LAYERS; ++l) {
    // ---- phase 1: LN -> rotor sandwich -> geometric GELU : shCur -> shTmp ----
    {
      // prefetch next layer's 128KB weight block into cache while we do VALU work
      if (l + 1 < LAYERS) {
        const char* nw = (const char*)(lin_W + (size_t)(l+1)*DIM*HID*HID);
        __builtin_prefetch(nw + tid*256,       0, 2);
        __builtin_prefetch(nw + tid*256 + 128, 0, 2);
      }
      const int b  = tid & 63;
      const int cg = tid >> 6;
      for (int ci = 0; ci < 8; ++ci) {
        const int c  = cg*8 + ci;
        const int lc = l*64 + c;
        float h[8]; float ss = 0.0f;
        #pragma unroll
        for (int k = 0; k < 8; ++k) { h[k] = shCur[IDXH(k,c,b)]; ss += h[k]*h[k]; }
        const float inv = ln_g[lc] * rsqrtf(ss*0.125f + 1e-6f);
        #pragma unroll
        for (int k = 0; k < 8; ++k) h[k] *= inv;
        // rotor R = cos(t/2) - sin(t/2)*Bhat
        const float bx = rotor_biv[lc*3+0], by = rotor_biv[lc*3+1], bz = rotor_biv[lc*3+2];
        const float th = sqrtf(bx*bx + by*by + bz*bz + 1e-12f);
        const float sc = sinf(0.5f*th) / th;
        const float r0 = cosf(0.5f*th);
        const float R4 = -sc*bx, R5 = -sc*by, R6 = -sc*bz;
        float rh[8], g2[8];
        gp_rotor_left(r0, R4, R5, R6, h, rh);
        gp_rotor_right(rh, r0, -R4, -R5, -R6, g2);     // Rrev flips bivector signs
        // geometric GELU gate
        float s2 = 0.0f;
        #pragma unroll
        for (int k = 0; k < 8; ++k) s2 += g2[k]*g2[k];
        const float x    = act_w[lc]*sqrtf(s2 + 1e-9f) + act_b[lc];
        const float gate = 0.5f*(1.0f + erff(x*0.70710678118654752f));
        #pragma unroll
        for (int k = 0; k < 8; ++k) shTmp[IDXH(k,c,b)] = g2[k]*gate;
      }
    }
    // previous layer's async inter-copy must be done before phase 2 rewrites shCur;
    // wait on our own async stores, then barrier orders it across waves.
    asm volatile("s_wait_asynccnt 0x0" ::: "memory");
    __syncthreads();

    // ---- phase 2: blade-wise 64x64 GEMM + bias + residual (fp32 WMMA) ----
    {
      const int k  = wave & 7;             // blade owned by this wave
      const int mh = wave >> 3;            // M half: rows [mh*32, mh*32+32)
      const float* Wl = lin_W + (size_t)((l*8 + k)*64)*64;   // [o][i]
      for (int nt = 0; nt < 4; ++nt) {
        const int o = nt*16 + ln15;
        v2f bfr[16];                        // B fragments held across both M tiles
        #pragma unroll
        for (int kk = 0; kk < 16; ++kk) {
          const int i0 = kk*4 + 2*hi;
          const float* wp = Wl + (size_t)o*64 + i0;
          bfr[kk].x = wp[0]; bfr[kk].y = wp[1];
        }
        const float bias = lin_b[(l*64 + o)*8 + k];
        for (int mq = 0; mq < 2; ++mq) {
          const int mt = mh*2 + mq;
          v8f acc;
          #pragma unroll
          for (int r = 0; r < 8; ++r) {    // init = residual + bias
            const int b = mt*16 + r + 8*hi;
            acc[r] = shCur[IDXH(k,o,b)] + bias;
          }
          const int bA = mt*16 + ln15;
          #pragma unroll
          for (int kk = 0; kk < 16; ++kk) {
            const int i0 = kk*4 + 2*hi;
            v2f a;
            a.x = shTmp[IDXH(k, i0,     bA)];
            a.y = shTmp[IDXH(k, i0 + 1, bA)];
            acc = __builtin_amdgcn_wmma_f32_16x16x4_f32(false, a, false, bfr[kk],
                                                        (short)0, acc, false, false);
          }
          #pragma unroll
          for (int r = 0; r < 8; ++r) {
            const int b = mt*16 + r + 8*hi;
            shCur[IDXH(k,o,b)] = acc[r];
          }
        }
      }
    }
    __syncthreads();

    // ---- phase 3: inter[l] dump via async LDS->global stores (non-temporal) ----
    // lanes contiguous over j = c*8+k -> contiguous global addresses
    {
      const int c = tid >> 3;
      const int k = tid & 7;
      float* basep = out_inter + ((size_t)l*NB + B0)*512;
      const unsigned lds0 = (unsigned)(uintptr_t)&shCur[IDXH(k,c,0)];
      for (int b = 0; b < BT; ++b) {
        const unsigned voff = (unsigned)((b*512 + tid)*4);
        const unsigned lof  = lds0 + (unsigned)(b*4);
        asm volatile("global_store_async_from_lds_b32 %0, %1, %2 th:TH_STORE_NT"
                     :: "v"(voff), "v"(lof), "s"(basep) : "memory");
      }
    }
    // no barrier needed: next phase 1 only reads shCur / writes shTmp
  }

  // ---- final: out-LN + blade gate + projection ----
  {
    const int b = tid & 63;
    const int g = tid >> 6;
    float p[8];
    #pragma unroll
    for (int k = 0; k < 8; ++k) p[k] = 0.0f;
    for (int ci = 0; ci < 8; ++ci) {
      const int c = g*8 + ci;
      float v[8], ss = 0.0f;
      #pragma unroll
      for (int k = 0; k < 8; ++k) { v[k] = shCur[IDXH(k,c,b)]; ss += v[k]*v[k]; }
      const float inv = out_ln_g[c] * rsqrtf(ss*0.125f + 1e-6f);
      #pragma unroll
      for (int k = 0; k < 8; ++k) {
        const float gate = 1.0f/(1.0f + expf(-blade_sel[c*8 + k]));
        p[k] += v[k]*inv*gate*proj_W[k*64 + c];
      }
    }
    #pragma unroll
    for (int k = 0; k < 8; ++k) shTmp[(g*64 + b)*8 + k] = p[k];
  }
  __syncthreads();
  {
    const int b = tid >> 3, k = tid & 7;
    float s = proj_b[k];
    #pragma unroll
    for (int g = 0; g < 8; ++g) s += shTmp[(g*64 + b)*8 + k];
    out_final[((size_t)B0 + b)*8 + k] = s;
  }
  // kernel end: S_ENDPGM implicitly waits for all outstanding (incl. ASYNCcnt)
}

// ---------------- launch ----------------
extern "C" void kernel_launch(void* const* d_in, const int* in_sizes, int n_in,
                              void* d_out, int out_size, void* d_ws, size_t ws_size,
                              hipStream_t stream) {
  // setup_inputs order: coords, temporal_freqs, re_freqs, then params dict flattened:
  // lift_W, lift_b, in_ln_g, ln_g, rotor_biv, act_w, act_b, lin_W, lin_b,
  // out_ln_g, blade_sel, proj_W, proj_b
  const float* coords    = (const float*)d_in[0];
  const float* tfreq     = (const float*)d_in[1];
  const float* rfreq     = (const float*)d_in[2];
  const float* lift_W    = (const float*)d_in[3];
  const float* lift_b    = (const float*)d_in[4];
  const float* in_ln_g   = (const float*)d_in[5];
  const float* ln_g      = (const float*)d_in[6];
  const float* rotor_biv = (const float*)d_in[7];
  const float* act_w     = (const float*)d_in[8];
  const float* act_b     = (const float*)d_in[9];
  const float* lin_W     = (const float*)d_in[10];
  const float* lin_b     = (const float*)d_in[11];
  const float* out_ln_g  = (const float*)d_in[12];
  const float* blade_sel = (const float*)d_in[13];
  const float* proj_W    = (const float*)d_in[14];
  const float* proj_b    = (const float*)d_in[15];

  float* out_final = (float*)d_out;                 // [NB,8]
  float* out_inter = out_final + (size_t)NB*8;      // [6,NB,64,8]

  gaugefluid_fused<<<dim3(NB/BT), dim3(NTHREADS), 0, stream>>>(
      coords, tfreq, rfreq, lift_W, lift_b, in_ln_g, ln_g, rotor_biv,
      act_w, act_b, lin_W, lin_b, out_ln_g, blade_sel, proj_W, proj_b,
      out_final, out_inter);
}